// CustomNPairsLoss_910533066837
// MI455X (gfx1250) — compile-verified
//
#include <hip/hip_runtime.h>
#include <hip/hip_bf16.h>

typedef float v2f __attribute__((ext_vector_type(2)));
typedef float v8f __attribute__((ext_vector_type(8)));

#define EMB_D    64
#define TILE     128
#define LDSPITCH 68   // 68 floats = 272B row pitch: 16B-aligned, breaks 64-bank conflicts
#define MARGIN   1.0f
#define EPSA     1e-6f

// ---------------------------------------------------------------------------
// Phase 1: per-row norms, diagonal distance, label mask. One wave32 per row.
// ---------------------------------------------------------------------------
__global__ __launch_bounds__(256) void npl_prep(const float* __restrict__ A,
                                                const float* __restrict__ P,
                                                const int* __restrict__ labels,
                                                float* __restrict__ a2,
                                                float* __restrict__ p2,
                                                float* __restrict__ posd,
                                                float* __restrict__ lmask,
                                                int N) {
    int row  = (blockIdx.x * blockDim.x + threadIdx.x) >> 5;
    int lane = threadIdx.x & 31;
    if (row >= N) return;
    const float* ar = A + (size_t)row * EMB_D;
    const float* pr = P + (size_t)row * EMB_D;
    float ax0 = ar[lane] + EPSA, ax1 = ar[lane + 32] + EPSA;
    float px0 = pr[lane],        px1 = pr[lane + 32];
    float sa = ax0 * ax0 + ax1 * ax1;
    float sp = px0 * px0 + px1 * px1;
    float sd = ax0 * px0 + ax1 * px1;
    for (int off = 16; off > 0; off >>= 1) {
        sa += __shfl_xor(sa, off, 32);
        sp += __shfl_xor(sp, off, 32);
        sd += __shfl_xor(sd, off, 32);
    }
    if (lane == 0) {
        a2[row] = sa;
        p2[row] = sp;
        float sq = sa + sp - 2.0f * sd;
        posd[row]  = __builtin_amdgcn_sqrtf(fmaxf(sq, 1e-12f));
        lmask[row] = (labels[row] == 1) ? 1.0f : 0.0f;
    }
}

// ---------------------------------------------------------------------------
// Phase 2: fused WMMA-f32 Gram tile + distance/hinge epilogue + block reduce.
// Block = 8 wave32s, 128x128 output tile. Wave w owns rows [i0+16w, i0+16w+16).
// Diagonal (i==j) terms are INCLUDED here (each contributes lmask[i]*MARGIN
// analytically) and subtracted once in the finalize kernel — this removes all
// per-element index compares from the hot loop.
// ---------------------------------------------------------------------------
__global__ __launch_bounds__(256) void npl_tile(const float* __restrict__ A,
                                                const float* __restrict__ P,
                                                const float* __restrict__ a2,
                                                const float* __restrict__ p2,
                                                const float* __restrict__ posd,
                                                const float* __restrict__ lmask,
                                                float* __restrict__ partials,
                                                int N) {
    __shared__ float lds[TILE * LDSPITCH];

    const int tid  = threadIdx.x;
    const int lane = tid & 31;
    const int wave = tid >> 5;
    const int i0   = blockIdx.y * TILE;
    const int j0   = blockIdx.x * TILE;

    // Stage P tile [128 rows x 64 floats] -> LDS via async 16B copies.
    for (int t = tid; t < TILE * (EMB_D / 4); t += 256) {
        int r = t >> 4;           // row within tile
        int c = t & 15;           // float4 column
        const float* gsrc = P + (size_t)(j0 + r) * EMB_D + c * 4;
        unsigned lds_off = (unsigned)(size_t)&lds[r * LDSPITCH + c * 4];
        asm volatile("global_load_async_to_lds_b128 %0, %1, off"
                     :: "v"(lds_off), "v"(gsrc) : "memory");
    }

    // A fragments (with +EPS) for this wave's 16-row strip, all K in registers.
    // A 16x4 f32 layout: lanes 0-15 -> K = k0+{0,1}; lanes 16-31 -> K = k0+{2,3}.
    const int m  = lane & 15;
    const int kk = (lane >> 4) * 2;
    const int irow = i0 + wave * 16 + m;
    v2f afrag[16];
#pragma unroll
    for (int k = 0; k < 16; ++k) {
        afrag[k].x = A[(size_t)irow * EMB_D + k * 4 + kk]     + EPSA;
        afrag[k].y = A[(size_t)irow * EMB_D + k * 4 + kk + 1] + EPSA;
    }

    // Per-lane epilogue constants for the 8 M-rows this lane will hold.
    // C/D layout: lane n=lane&15; VGPR r holds M = r + (lane>=16 ? 8 : 0).
    const int mbase = (lane >> 4) * 8;
    float ia2[8], ipd1[8], ilm[8];
#pragma unroll
    for (int r = 0; r < 8; ++r) {
        int i = i0 + wave * 16 + mbase + r;
        ia2[r]  = a2[i];
        ipd1[r] = posd[i] + MARGIN;
        ilm[r]  = lmask[i];
    }
    // Hoist all 8 column-norm loads out of the main loop.
    float pj2[8];
#pragma unroll
    for (int jt = 0; jt < 8; ++jt)
        pj2[jt] = p2[j0 + jt * 16 + m];

    // Drain async LDS writes, then block barrier before any LDS reads.
    asm volatile("s_wait_asynccnt 0x0" ::: "memory");
    __syncthreads();

    float lsum = 0.0f;
    unsigned boff = (unsigned)(m * LDSPITCH + kk);   // running LDS row base
#pragma unroll 1
    for (int jt = 0; jt < 8; ++jt) {
        // Batch all 16 B-fragment LDS loads.
        v2f bfrag[16];
#pragma unroll
        for (int k = 0; k < 16; ++k)
            bfrag[k] = *(const v2f*)&lds[boff + k * 4];

        // Register wall: every ds_load must land before this point, every
        // WMMA consumes its outputs -> 16 loads, ONE dscnt wait, 16-WMMA burst.
        asm volatile(""
            : "+v"(bfrag[0]),  "+v"(bfrag[1]),  "+v"(bfrag[2]),  "+v"(bfrag[3]),
              "+v"(bfrag[4]),  "+v"(bfrag[5]),  "+v"(bfrag[6]),  "+v"(bfrag[7]),
              "+v"(bfrag[8]),  "+v"(bfrag[9]),  "+v"(bfrag[10]), "+v"(bfrag[11]),
              "+v"(bfrag[12]), "+v"(bfrag[13]), "+v"(bfrag[14]), "+v"(bfrag[15]));

        v8f acc = {0.f, 0.f, 0.f, 0.f, 0.f, 0.f, 0.f, 0.f};
#pragma unroll
        for (int k = 0; k < 16; ++k)
            acc = __builtin_amdgcn_wmma_f32_16x16x4_f32(
                false, afrag[k], false, bfrag[k], (short)0, acc, false, false);

        const float pj = pj2[jt];
#pragma unroll
        for (int r = 0; r < 8; ++r) {
            float sq   = ia2[r] + pj - 2.0f * acc[r];
            float dist = __builtin_amdgcn_sqrtf(fmaxf(sq, 1e-12f));
            float h    = fmaxf(ipd1[r] - dist, 0.0f);
            lsum = fmaf(h, ilm[r], lsum);
        }

        boff += 16u * LDSPITCH;
    }

    // Wave reduce, then cross-wave via LDS (after all waves finish LDS reads).
    for (int off = 16; off > 0; off >>= 1)
        lsum += __shfl_xor(lsum, off, 32);
    __syncthreads();
    if (lane == 0) lds[wave] = lsum;
    __syncthreads();
    if (tid == 0) {
        float s = 0.0f;
        for (int w = 0; w < 8; ++w) s += lds[w];
        partials[blockIdx.y * gridDim.x + blockIdx.x] = s;
    }
}

// ---------------------------------------------------------------------------
// Phase 3: deterministic final reduction (fixed strided order + fixed tree).
// Subtracts the analytic diagonal contribution cnt1*MARGIN.
// ---------------------------------------------------------------------------
__global__ __launch_bounds__(256) void npl_finalize(const float* __restrict__ partials,
                                                    int nPart,
                                                    const float* __restrict__ lmask,
                                                    int N,
                                                    float* __restrict__ out) {
    __shared__ float red[256];
    const int tid = threadIdx.x;

    float s = 0.0f;
    for (int t = tid; t < nPart; t += 256) s += partials[t];
    red[tid] = s;
    __syncthreads();
    for (int off = 128; off > 0; off >>= 1) {
        if (tid < off) red[tid] += red[tid + off];
        __syncthreads();
    }
    float total = red[0];
    __syncthreads();

    float c = 0.0f;
    for (int t = tid; t < N; t += 256) c += lmask[t];
    red[tid] = c;
    __syncthreads();
    for (int off = 128; off > 0; off >>= 1) {
        if (tid < off) red[tid] += red[tid + off];
        __syncthreads();
    }
    if (tid == 0) {
        float cnt1  = red[0];
        float count = cnt1 * (float)(N - 1);
        float num   = total - cnt1 * MARGIN;   // remove diagonal terms
        out[0] = (count > 0.0f) ? num / fmaxf(count, 1.0f) : 0.0f;
    }
}

// ---------------------------------------------------------------------------
extern "C" void kernel_launch(void* const* d_in, const int* in_sizes, int n_in,
                              void* d_out, int out_size, void* d_ws, size_t ws_size,
                              hipStream_t stream) {
    const float* A      = (const float*)d_in[0];
    const float* P      = (const float*)d_in[1];
    const int*   labels = (const int*)d_in[2];
    float*       out    = (float*)d_out;

    const int N = in_sizes[2];          // 8192
    float* ws = (float*)d_ws;
    float* a2       = ws;
    float* p2       = ws + N;
    float* posd     = ws + 2 * N;
    float* lmask    = ws + 3 * N;
    float* partials = ws + 4 * N;

    const int nTiles = N / TILE;        // 64

    npl_prep<<<(N + 7) / 8, 256, 0, stream>>>(A, P, labels, a2, p2, posd, lmask, N);

    dim3 grid(nTiles, nTiles);
    npl_tile<<<grid, 256, 0, stream>>>(A, P, a2, p2, posd, lmask, partials, N);

    npl_finalize<<<1, 256, 0, stream>>>(partials, nTiles * nTiles, lmask, N, out);
}